// Autocorrelation_66718021976429
// MI455X (gfx1250) — compile-verified
//
#include <hip/hip_runtime.h>
#include <hip/hip_bf16.h>

#define BATCH 4
#define SEQ_L 4096
#define D_MODEL 512
#define D_K 64
#define HEADS 8
#define TOPK 16
#define LMASK (SEQ_L - 1)

typedef float v2f __attribute__((ext_vector_type(2)));
typedef float v8f __attribute__((ext_vector_type(8)));

__device__ __forceinline__ v8f wmma_f32_4(v2f a, v2f b, v8f c) {
  // V_WMMA_F32_16X16X4_F32 : D = A(16x4) * B(4x16) + C(16x16)
  return __builtin_amdgcn_wmma_f32_16x16x4_f32(
      false, a, false, b, (short)0, c, false, false);
}

// Async global -> LDS copy (CDNA5 GLOBAL_LOAD_ASYNC_TO_LDS_B128, ASYNCcnt).
// lds_off = LDS byte offset (low 32 bits of the flat shared-aperture address).
__device__ __forceinline__ void async_copy_b128(unsigned lds_off, const void* gptr) {
  asm volatile("global_load_async_to_lds_b128 %0, %1, off"
               :: "v"(lds_off), "v"(gptr)
               : "memory");
}
__device__ __forceinline__ void wait_async0() {
  asm volatile("s_wait_asynccnt 0x0" ::: "memory");
}
__device__ __forceinline__ unsigned lds_offset(const void* p) {
  return (unsigned)(uintptr_t)p;   // shared aperture: low 32 bits = LDS offset
}

// ---------------------------------------------------------------------------
// Kernel 1: p = x @ Wq + bq for q/k/v, written transposed as [b][d][t].
// One wave computes a 16(t) x 64(d) stripe: single K-loop, A loaded once per
// chunk, 4 accumulators against the 4 W column tiles (A traffic / 4).
// 3 inputs * 1024 row-tiles = 3072 waves -> 384 blocks of 256 threads.
// ---------------------------------------------------------------------------
__global__ __launch_bounds__(256) void proj_kernel(
    const float* __restrict__ q_in, const float* __restrict__ k_in,
    const float* __restrict__ v_in, const float* __restrict__ W,
    const float* __restrict__ bias, float* __restrict__ pq,
    float* __restrict__ pk, float* __restrict__ pv) {
  const int wave = blockIdx.x * 8 + (threadIdx.x >> 5);
  const int lane = threadIdx.x & 31;
  const int inp  = wave >> 10;      // 0=q 1=k 2=v
  const int rt   = wave & 1023;     // tile over B*L/16 rows
  const int row0 = rt << 4;
  const int b    = row0 >> 12;      // / SEQ_L
  const int t0   = row0 & LMASK;

  const float* x    = (inp == 0) ? q_in : (inp == 1) ? k_in : v_in;
  float*       pout = (inp == 0) ? pq   : (inp == 1) ? pk   : pv;

  const bool hi  = lane >= 16;
  const int  l16 = lane & 15;
  const int  u0  = hi ? 2 : 0;      // K sub-offset per documented A layout
  const float* arow = x + (size_t)(row0 + l16) * D_MODEL;

  v8f acc0 = (v8f)(0.0f), acc1 = (v8f)(0.0f);
  v8f acc2 = (v8f)(0.0f), acc3 = (v8f)(0.0f);
#pragma unroll 4
  for (int c4 = 0; c4 < D_MODEL; c4 += 4) {
    v2f a;
    a.x = arow[c4 + u0];
    a.y = arow[c4 + u0 + 1];
    const float* w0 = W + (c4 + u0) * D_K + l16;       // row u0
    const float* w1 = w0 + D_K;                        // row u0+1
    v2f b0, b1, b2, b3;
    b0.x = w0[0];  b0.y = w1[0];
    b1.x = w0[16]; b1.y = w1[16];
    b2.x = w0[32]; b2.y = w1[32];
    b3.x = w0[48]; b3.y = w1[48];
    acc0 = wmma_f32_4(a, b0, acc0);
    acc1 = wmma_f32_4(a, b1, acc1);
    acc2 = wmma_f32_4(a, b2, acc2);
    acc3 = wmma_f32_4(a, b3, acc3);
  }
  const int mrow = hi ? 8 : 0;
#pragma unroll
  for (int nt = 0; nt < 4; ++nt) {
    const v8f acc = (nt == 0) ? acc0 : (nt == 1) ? acc1 : (nt == 2) ? acc2 : acc3;
    const int d0 = nt << 4;
    const float bv = bias[d0 + l16];
    float* col = pout + ((size_t)(b * D_K + d0 + l16) << 12) + t0 + mrow;
#pragma unroll
    for (int r = 0; r < 8; ++r) col[r] = acc[r] + bv;
  }
}

// ---------------------------------------------------------------------------
// Kernel 2: per (b,d) row: circular correlation via WMMA from LDS,
// |.|, block-cooperative top-16 (stable ties -> lowest index), softmax.
// Rows staged into LDS with async global->LDS B128 copies (no VGPR bounce).
// 8 waves per row; wave w owns lag supertiles [256w, 256w+255] and
// [2048+256w, 2048+256w+255] (shared A-operand load, 2 accumulators).
// ---------------------------------------------------------------------------
__global__ __launch_bounds__(256) void corr_topk_kernel(
    const float* __restrict__ pq, const float* __restrict__ pk,
    float* __restrict__ wtop, int* __restrict__ itop) {
  __shared__ float qs[SEQ_L];
  __shared__ float ks[SEQ_L + 4];   // head duplicated: ks[L..L+3] = k[0..3]
  __shared__ float cabs[SEQ_L];
  __shared__ float rv[8];
  __shared__ int   ri[8];
  __shared__ float topv[TOPK];
  __shared__ int   topi[TOPK];

  const int row = blockIdx.x;               // b*64 + d
  const int tid = threadIdx.x;
  const float* qrow = pq + ((size_t)row << 12);
  const float* krow = pk + ((size_t)row << 12);

  // 16 B per lane per pass; 4 passes cover each 16 KB row.
  {
    const unsigned ql = lds_offset(qs) + tid * 16;
    const unsigned kl = lds_offset(ks) + tid * 16;
    const char* qg = (const char*)qrow + tid * 16;
    const char* kg = (const char*)krow + tid * 16;
#pragma unroll
    for (int p = 0; p < 4; ++p) {
      async_copy_b128(ql + p * 4096, qg + p * 4096);
      async_copy_b128(kl + p * 4096, kg + p * 4096);
    }
  }
  if (tid < 4) ks[SEQ_L + tid] = krow[tid];   // wrap duplicate (plain path)
  wait_async0();
  __syncthreads();

  const int  wave = tid >> 5;               // 0..7
  const int  lane = tid & 31;
  const bool hi   = lane >= 16;
  const int  l16  = lane & 15;
  const int  u0   = hi ? 2 : 0;
  const int  tau0 = wave << 8;
  const int  abase = (l16 << 4) + u0;                 // 16m + u
  const int  bb0   = u0 - tau0 - l16 + 2 * SEQ_L;     // stays non-negative
  const int  bb1   = bb0 - 2048;                      // supertile tau0+2048

  v8f acc0 = (v8f)(0.0f), acc1 = (v8f)(0.0f);
  for (int c4 = 0; c4 < SEQ_L; c4 += 4) {
    const int ai = (c4 + abase) & LMASK;              // even -> ai+1 in range
    v2f a;
    a.x = qs[ai];
    a.y = qs[ai + 1];
    const int bi0 = (c4 + bb0) & LMASK;               // <=4095 -> +1 safe
    v2f b0;
    b0.x = ks[bi0];
    b0.y = ks[bi0 + 1];
    const int bi1 = (c4 + bb1) & LMASK;
    v2f b1;
    b1.x = ks[bi1];
    b1.y = ks[bi1 + 1];
    acc0 = wmma_f32_4(a, b0, acc0);
    acc1 = wmma_f32_4(a, b1, acc1);
  }
  const int mb = hi ? 8 : 0;
#pragma unroll
  for (int r = 0; r < 8; ++r) {
    cabs[tau0 + ((r + mb) << 4) + l16]        = fabsf(acc0[r]);
    cabs[tau0 + 2048 + ((r + mb) << 4) + l16] = fabsf(acc1[r]);
  }
  __syncthreads();

  // ---- top-16, deterministic (first occurrence wins on ties) ----
  for (int it = 0; it < TOPK; ++it) {
    float bv = -1.0f; int bidx = 0;
    for (int i = tid; i < SEQ_L; i += 256) {
      const float v = cabs[i];
      if (v > bv) { bv = v; bidx = i; }               // ascending i -> stable
    }
    for (int off = 16; off > 0; off >>= 1) {
      const float ov = __shfl_xor(bv, off, 32);
      const int   oi = __shfl_xor(bidx, off, 32);
      if (ov > bv || (ov == bv && oi < bidx)) { bv = ov; bidx = oi; }
    }
    if (lane == 0) { rv[wave] = bv; ri[wave] = bidx; }
    __syncthreads();
    if (tid == 0) {
      float fv = rv[0]; int fi = ri[0];
      for (int w2 = 1; w2 < 8; ++w2) {
        const float v = rv[w2]; const int ix = ri[w2];
        if (v > fv || (v == fv && ix < fi)) { fv = v; fi = ix; }
      }
      topv[it] = fv; topi[it] = fi;
      cabs[fi] = -1.0f;                               // remove winner
    }
    __syncthreads();
  }

  if (tid == 0) {
    float mx = topv[0];
    for (int i = 1; i < TOPK; ++i) mx = fmaxf(mx, topv[i]);
    float e[TOPK]; float s = 0.0f;
    for (int i = 0; i < TOPK; ++i) { e[i] = expf(topv[i] - mx); s += e[i]; }
    const float inv = 1.0f / s;
    for (int i = 0; i < TOPK; ++i) {
      wtop[row * TOPK + i] = e[i] * inv;
      itop[row * TOPK + i] = topi[i];
    }
  }
}

// ---------------------------------------------------------------------------
// Kernel 3: agg[row][t] = sum_k w_k * v[row][(t+idx_k)&4095], v row in LDS
// (staged with async global->LDS copies).
// ---------------------------------------------------------------------------
__global__ __launch_bounds__(256) void agg_kernel(
    const float* __restrict__ pv, const float* __restrict__ wtop,
    const int* __restrict__ itop, float* __restrict__ agg) {
  __shared__ float vs[SEQ_L];
  __shared__ float wl[TOPK];
  __shared__ int   il[TOPK];
  const int row = blockIdx.x;
  const int tid = threadIdx.x;
  const float* vrow = pv + ((size_t)row << 12);
  {
    const unsigned vl = lds_offset(vs) + tid * 16;
    const char*    vg = (const char*)vrow + tid * 16;
#pragma unroll
    for (int p = 0; p < 4; ++p) async_copy_b128(vl + p * 4096, vg + p * 4096);
  }
  if (tid < TOPK) { wl[tid] = wtop[row * TOPK + tid]; il[tid] = itop[row * TOPK + tid]; }
  wait_async0();
  __syncthreads();
  float* arow = agg + ((size_t)row << 12);
  for (int t = tid; t < SEQ_L; t += 256) {
    float s = 0.0f;
#pragma unroll
    for (int k2 = 0; k2 < TOPK; ++k2)
      s = fmaf(wl[k2], vs[(t + il[k2]) & LMASK], s);
    arow[t] = s;
  }
}

// ---------------------------------------------------------------------------
// Kernel 4: out[b][t][h*64+d] = agg[b][d][t], LDS transpose, coalesced stores.
// Block handles one (b, 32-wide t-tile): loads 64x32, writes 32x512.
// ---------------------------------------------------------------------------
__global__ __launch_bounds__(256) void writeout_kernel(
    const float* __restrict__ agg, float* __restrict__ out) {
  __shared__ float tile[D_K][33];
  const int b  = blockIdx.x >> 7;               // 128 t-tiles per batch
  const int t0 = (blockIdx.x & 127) << 5;
  const int tid = threadIdx.x;
  const int tt  = tid & 31;
  const int db  = tid >> 5;                     // 0..7
#pragma unroll
  for (int dd = 0; dd < 8; ++dd) {
    const int d = dd * 8 + db;
    tile[d][tt] = agg[((size_t)(b * D_K + d) << 12) + t0 + tt];
  }
  __syncthreads();
#pragma unroll 4
  for (int it = 0; it < 64; ++it) {
    const int flat = it * 256 + tid;
    const int c    = flat & 511;                // h*64+d
    const int trel = flat >> 9;
    out[((size_t)(b * SEQ_L + t0 + trel) << 9) + c] = tile[c & 63][trel];
  }
}

// ---------------------------------------------------------------------------
extern "C" void kernel_launch(void* const* d_in, const int* in_sizes, int n_in,
                              void* d_out, int out_size, void* d_ws, size_t ws_size,
                              hipStream_t stream) {
  (void)in_sizes; (void)n_in; (void)out_size; (void)ws_size;
  const float* q_in = (const float*)d_in[0];
  const float* k_in = (const float*)d_in[1];
  const float* v_in = (const float*)d_in[2];
  const float* Wq   = (const float*)d_in[3];
  const float* bq   = (const float*)d_in[4];
  float* out = (float*)d_out;

  // workspace layout (floats): pq | pk | pv | wtop | itop | agg  (~16.8 MB)
  float* ws   = (float*)d_ws;
  const size_t PROW = (size_t)BATCH * D_K * SEQ_L;   // 1<<20
  float* pq   = ws;
  float* pk   = ws + PROW;
  float* pv   = ws + 2 * PROW;
  float* wtop = ws + 3 * PROW;
  int*   itop = (int*)(ws + 3 * PROW + BATCH * D_K * TOPK);
  float* agg  = ws + 3 * PROW + 2 * BATCH * D_K * TOPK;

  proj_kernel<<<384, 256, 0, stream>>>(q_in, k_in, v_in, Wq, bq, pq, pk, pv);
  corr_topk_kernel<<<BATCH * D_K, 256, 0, stream>>>(pq, pk, wtop, itop);
  agg_kernel<<<BATCH * D_K, 256, 0, stream>>>(pv, wtop, itop, agg);
  writeout_kernel<<<BATCH * 128, 256, 0, stream>>>(agg, out);
}